// SLLoss_st2vec_27144193311431
// MI455X (gfx1250) — compile-verified
//
#include <hip/hip_runtime.h>
#include <hip/hip_bf16.h>

typedef float v2f __attribute__((ext_vector_type(2)));
typedef float v8f __attribute__((ext_vector_type(8)));

#define B_ROWS 262144
#define D_DIM  128
#define WAVES_PER_BLOCK 8
#define ROWS_PER_WAVE   16
#define NUM_BLOCKS (B_ROWS / (ROWS_PER_WAVE * WAVES_PER_BLOCK))  // 2048

// Stage 1: each wave computes squared L2 distances for a 16-row tile via
// V_WMMA_F32_16X16X4_F32 with an all-ones B matrix (rowsum trick), then
// evaluates the per-sample loss and reduces to one partial per block.
__global__ __launch_bounds__(256) void sll_partial_kernel(
    const float* __restrict__ ea, const float* __restrict__ ep,
    const float* __restrict__ en, const float* __restrict__ gt,
    float* __restrict__ partial)
{
    const int lane        = threadIdx.x & 31;
    const int waveInBlock = threadIdx.x >> 5;
    const int waveGlobal  = blockIdx.x * WAVES_PER_BLOCK + waveInBlock;
    const int tileRow     = waveGlobal * ROWS_PER_WAVE;

    // A-operand layout for 16x16x4 f32 WMMA:
    //   lanes 0-15  : row = lane,     VGPR0/1 = K=k, K=k+1
    //   lanes 16-31 : row = lane-16,  VGPR0/1 = K=k+2, K=k+3
    const int rowInTile = lane & 15;
    const int kHalf     = (lane >> 4) * 2;      // 0 or 2
    const size_t rowBase = (size_t)(tileRow + rowInTile) * D_DIM + kHalf;

    v8f c_ap = {};
    v8f c_an = {};
    const v2f ones = {1.0f, 1.0f};

    #pragma unroll 4
    for (int k = 0; k < D_DIM; k += 4) {
        const size_t off = rowBase + (size_t)k;
        v2f a2 = *(const v2f*)(ea + off);
        v2f p2 = *(const v2f*)(ep + off);
        v2f n2 = *(const v2f*)(en + off);
        v2f dap = a2 - p2;
        v2f dan = a2 - n2;
        v2f sap = dap * dap;
        v2f san = dan * dan;
        // D[m,n] += sum_k sap[m,k] * 1.0  -> every column of row m holds ||a-p||^2
        c_ap = __builtin_amdgcn_wmma_f32_16x16x4_f32(
            false, sap, false, ones, (short)0, c_ap, false, false);
        c_an = __builtin_amdgcn_wmma_f32_16x16x4_f32(
            false, san, false, ones, (short)0, c_an, false, false);
    }

    // C/D layout: lanes 0-15 VGPR i = row i; lanes 16-31 VGPR i = row i+8.
    // Lane 0 owns rows tile+0..7, lane 16 owns rows tile+8..15 (all N equal).
    float acc = 0.0f;
    if ((lane & 15) == 0) {
        const int rbase = tileRow + ((lane >> 4) << 3);
        #pragma unroll
        for (int i = 0; i < 8; ++i) {
            const int r = rbase + i;
            float g0 = gt[2 * r + 0];
            float g1 = gt[2 * r + 1];
            float Dap = expf(-g0);
            float Dan = expf(-g1);
            float vap = expf(-sqrtf(c_ap[i]));
            float van = expf(-sqrtf(c_an[i]));
            float tap = Dap - vap;
            float tan_ = Dan - van;
            acc += Dap * tap * tap + Dan * tan_ * tan_;
        }
    }

    // Wave reduction (wave32), then cross-wave via LDS. Fixed order -> deterministic.
    #pragma unroll
    for (int m = 16; m >= 1; m >>= 1)
        acc += __shfl_xor(acc, m, 32);

    __shared__ float smem[WAVES_PER_BLOCK];
    if (lane == 0) smem[waveInBlock] = acc;
    __syncthreads();
    if (threadIdx.x == 0) {
        float s = 0.0f;
        #pragma unroll
        for (int i = 0; i < WAVES_PER_BLOCK; ++i) s += smem[i];
        partial[blockIdx.x] = s;
    }
}

// Stage 2: single block reduces the per-block partials in a fixed order
// and writes mean = sum / B.
__global__ __launch_bounds__(256) void sll_final_kernel(
    const float* __restrict__ partial, float* __restrict__ out, int n)
{
    __shared__ float smem[256];
    float s = 0.0f;
    for (int i = threadIdx.x; i < n; i += 256) s += partial[i];
    smem[threadIdx.x] = s;
    __syncthreads();
    #pragma unroll
    for (int step = 128; step >= 1; step >>= 1) {
        if ((int)threadIdx.x < step) smem[threadIdx.x] += smem[threadIdx.x + step];
        __syncthreads();
    }
    if (threadIdx.x == 0) out[0] = smem[0] * (1.0f / (float)B_ROWS);
}

extern "C" void kernel_launch(void* const* d_in, const int* in_sizes, int n_in,
                              void* d_out, int out_size, void* d_ws, size_t ws_size,
                              hipStream_t stream) {
    const float* ea = (const float*)d_in[0];
    const float* ep = (const float*)d_in[1];
    const float* en = (const float*)d_in[2];
    const float* gt = (const float*)d_in[3];
    float* out      = (float*)d_out;
    float* partial  = (float*)d_ws;   // NUM_BLOCKS floats = 8 KB scratch

    sll_partial_kernel<<<NUM_BLOCKS, 256, 0, stream>>>(ea, ep, en, gt, partial);
    sll_final_kernel<<<1, 256, 0, stream>>>(partial, out, NUM_BLOCKS);
}